// bscnet_65249143161428
// MI455X (gfx1250) — compile-verified
//
#include <hip/hip_runtime.h>
#include <stdint.h>

#define NIMG 8
#define FULL 2048
#define DS   256          // downsampled dim (2048/8)
#define SC   8            // SCALE
#define HSTR 257          // H tile row stride: bank = (t + c) % 64, conflict-free
#define VSTR 33           // V tile row stride: bank = (33c + t) % 64, conflict-free
#define TILE_FLOATS 8448  // max(32*257=8224, 256*33=8448) -> 33 KB static LDS

typedef float v4f __attribute__((ext_vector_type(4)));

// ---------- CDNA5 (gfx1250) async global->LDS (ASYNCcnt-tracked) ----------
// The LDS object's address is passed as an (unused) asm input so it is CAPTURED:
// the "memory" clobber then legally covers the tile, preventing LLVM from folding
// the (asm-written) LDS reads to poison. Single static __shared__ object -> LDS
// offset 0, so the manual byte offset matches the hardware VDST LDS address.
__device__ __forceinline__ void async_ld_f32(const float* g, uint32_t lds_byte_off,
                                             const float* lds_obj) {
    asm volatile("global_load_async_to_lds_b32 %0, %1, off"
                 :: "v"(lds_byte_off),
                    "v"((uint64_t)(uintptr_t)g),
                    "v"((uint64_t)(uintptr_t)lds_obj)   // capture LDS object
                 : "memory");
}
__device__ __forceinline__ void wait_async0() {
    asm volatile("s_wait_asynccnt 0x0" ::: "memory");
}

// One wave (32 threads) per block; thread owns one full row (H) or column (V).
template <bool VERT>
__device__ __forceinline__ void scan_path(float* tile,
                                          const float* __restrict__ x0,
                                          const float* __restrict__ x1,
                                          float* __restrict__ ws_lr,
                                          float* __restrict__ ws_s)
{
    const int t   = threadIdx.x;          // 0..31
    const int grp = blockIdx.x;           // 0..7
    const int n   = blockIdx.y;           // image
    const int ent = grp * 32 + t;         // my row (H) or column (V), 0..255

    const float* img0 = x0 + (size_t)n * FULL * FULL;
    const float* img1 = x1 + (size_t)n * FULL * FULL;

    // ---- marker bitmask for my row/column, built in registers from x1 ----
    unsigned myw[8];
    #pragma unroll
    for (int w = 0; w < 8; ++w) {
        unsigned bits = 0u;
        #pragma unroll 4
        for (int b = 0; b < 32; ++b) {
            const int p = w * 32 + b;     // position along my line
            const size_t off = VERT ? ((size_t)(p * SC) * FULL + (size_t)ent * SC)
                                    : ((size_t)(ent * SC) * FULL + (size_t)p * SC);
            bits |= ((img1[off] == 0.0f) ? 1u : 0u) << b;
        }
        myw[w] = bits;
    }

    // ---- async-stage x0 strided-gather into LDS tile ----
    if (VERT) {
        // tile[r*33 + t] = img0[(r*8)*FULL + ent*8]   (lane stride 32B)
        #pragma unroll 4
        for (int r = 0; r < DS; ++r)
            async_ld_f32(img0 + (size_t)(r * SC) * FULL + (size_t)ent * SC,
                         (uint32_t)((r * VSTR + t) * 4), tile);
    } else {
        // tile[lr*257 + c] = img0[((grp*32+lr)*8)*FULL + c*8]   (lane stride 32B)
        #pragma unroll 1
        for (int lr = 0; lr < 32; ++lr) {
            const float* r0 = img0 + (size_t)((grp * 32 + lr) * SC) * FULL;
            #pragma unroll
            for (int cc = 0; cc < 8; ++cc) {
                const int c = cc * 32 + t;
                async_ld_f32(r0 + (size_t)c * SC, (uint32_t)((lr * HSTR + c) * 4), tile);
            }
        }
    }
    wait_async0();   // single-wave workgroup: no barrier needed

    float* lrb = ws_lr + (size_t)n * DS * DS;   // [pos][entity]  (coalesced)
    float* sb  = ws_s  + (size_t)n * DS * DS;   // row-major [row][col]

    unsigned flags[8];   // bit p: (marker exists strictly before p) && !marker(p)

    // ---- forward: left-to-right running means (left marker's x included) ----
    {
        bool seen = false; float sum = 0.0f, cnt = 0.0f;
        #pragma unroll
        for (int w = 0; w < 8; ++w) {
            unsigned f = 0u; const unsigned mw = myw[w];
            #pragma unroll 4
            for (int b = 0; b < 32; ++b) {
                const int p = w * 32 + b;
                const float x = VERT ? tile[p * VSTR + t] : tile[t * HSTR + p];
                const bool z = ((mw >> b) & 1u) != 0u;
                sum += x; cnt += 1.0f;                 // dead until first marker
                const bool lv = seen && !z;
                const float lr_ = lv ? sum * __builtin_amdgcn_rcpf(cnt) : x;
                if (lv) f |= (1u << b);
                lrb[p * DS + ent] = lr_;
                if (z) { seen = true; sum = x; cnt = 1.0f; }
            }
            flags[w] = f;
        }
    }

    // ---- backward: right-to-left running means + combine ----
    {
        bool seen = false; float sum = 0.0f, cnt = 0.0f;
        #pragma unroll
        for (int w = 7; w >= 0; --w) {
            const unsigned f = flags[w]; const unsigned mw = myw[w];
            #pragma unroll 4
            for (int b = 31; b >= 0; --b) {
                const int p = w * 32 + b;
                const float x = VERT ? tile[p * VSTR + t] : tile[t * HSTR + p];
                const bool z = ((mw >> b) & 1u) != 0u;
                sum += x; cnt += 1.0f;
                const bool interior = seen && !z && (((f >> b) & 1u) != 0u);
                const float rl_ = sum * __builtin_amdgcn_rcpf(cnt);
                const float val = interior ? (lrb[p * DS + ent] + rl_) : 2.0f * x;
                if (VERT) sb[p * DS + ent] = val;      // (row=p, col=ent), coalesced
                else      sb[ent * DS + p] = val;      // (row=ent, col=p)
                if (z) { seen = true; sum = x; cnt = 1.0f; }
            }
        }
    }
}

// grid = (8 groups, NIMG, 2 directions), block = 32 (one wave32)
__launch_bounds__(32)
__global__ void seg_scan_kernel(const float* __restrict__ x0,
                                const float* __restrict__ x1,
                                float* __restrict__ lrH, float* __restrict__ lrV,
                                float* __restrict__ sH,  float* __restrict__ sV)
{
    __shared__ float tile[TILE_FLOATS];   // single static LDS object -> offset 0
    if (blockIdx.z == 0) scan_path<false>(tile, x0, x1, lrH, sH);
    else                 scan_path<true >(tile, x0, x1, lrV, sV);
}

// 8x8 nearest-neighbor upsample of (sH + sV); each output float4 maps to one source elem.
__launch_bounds__(256)
__global__ void upsample8_kernel(const float* __restrict__ sH,
                                 const float* __restrict__ sV,
                                 v4f* __restrict__ out)
{
    const int gi  = blockIdx.x * 256 + threadIdx.x;   // 0 .. NIMG*FULL*(FULL/4)-1
    const int x4  = gi & (FULL / 4 - 1);              // 0..511
    const int row = gi >> 9;                          // n*FULL + y
    const int y   = row & (FULL - 1);
    const int n   = row >> 11;
    const size_t off = ((size_t)n * DS + (y >> 3)) * DS + (x4 >> 1);
    const float v = __ldg(sH + off) + __ldg(sV + off);
    const v4f v4 = {v, v, v, v};
    __builtin_nontemporal_store(v4, out + gi);        // 128 MB stream: don't retain in cache
}

extern "C" void kernel_launch(void* const* d_in, const int* in_sizes, int n_in,
                              void* d_out, int out_size, void* d_ws, size_t ws_size,
                              hipStream_t stream)
{
    (void)in_sizes; (void)n_in; (void)out_size; (void)ws_size;

    const float* x0 = (const float*)d_in[0];
    const float* x1 = (const float*)d_in[1];

    const size_t plane = (size_t)NIMG * DS * DS;      // 0.5M floats = 2 MB
    float* lrH = (float*)d_ws;
    float* lrV = lrH + plane;
    float* sH  = lrV + plane;
    float* sV  = sH  + plane;                         // total ws: 8 MB

    seg_scan_kernel<<<dim3(8, NIMG, 2), dim3(32), 0, stream>>>(x0, x1, lrH, lrV, sH, sV);

    const int total4 = NIMG * FULL * (FULL / 4);      // 8,388,608 float4 stores = 128 MB
    upsample8_kernel<<<dim3(total4 / 256), dim3(256), 0, stream>>>(sH, sV, (v4f*)d_out);
}